// CenterLoss_29575144800917
// MI455X (gfx1250) — compile-verified
//
#include <hip/hip_runtime.h>
#include <hip/hip_bf16.h>

// CenterLoss on MI455X (gfx1250).
//
// loss = mean(distmat * onehot(labels)) collapses to
//        (1/(B*C)) * sum_i ||x_i - centers[labels_i]||^2
// => pure gather + reduction (~4 MB traffic), NOT a GEMM. WMMA is
// deliberately unused: 21 GFLOP of the reference einsum is masked to zero.
// CDNA5-specific path exercised: async global->LDS gather pipeline
// (global_load_async_to_lds_b128 + s_wait_asynccnt double-buffering).

#define WAVES_PER_BLOCK 8
#define BLOCKS 64
#define THREADS (WAVES_PER_BLOCK * 32)
#define FEAT 128          // floats per row (reference FEAT_DIM)
#define F4 (FEAT / 4)     // 32 float4 per row -> one float4 per lane (wave32)

__global__ __launch_bounds__(THREADS)
void center_gather_partial(const float* __restrict__ x,
                           const int* __restrict__ labels,
                           const float* __restrict__ centers,
                           float* __restrict__ partials,
                           int batch, int spw)
{
    // Single shared object so our raw LDS byte offsets (base 0) used by the
    // async-to-LDS instruction line up with normal LDS reads of cbuf.
    // Per-wave double buffer: 2 x 32 lanes x 16B = 1 KB per wave (8 KB/block).
    __shared__ float4 cbuf[WAVES_PER_BLOCK * 2 * 32];

    const int lane = threadIdx.x & 31;
    const int wib  = threadIdx.x >> 5;                       // wave in block
    const int gw   = blockIdx.x * WAVES_PER_BLOCK + wib;     // global wave id
    const int s0   = gw * spw;
    int nv = batch - s0;
    if (nv > spw) nv = spw;

    float acc = 0.0f;
    if (nv > 0) {
        const float4* x4 = (const float4*)x;

        // Prologue: async-gather center row for sample 0 into buffer 0.
        {
            const int lbl = labels[s0];
            const float* gsrc = centers + (size_t)lbl * FEAT + lane * 4;
            unsigned ldsOff = (unsigned)(((wib * 2 + 0) * 32 + lane) * 16);
            asm volatile("global_load_async_to_lds_b128 %0, %1, off"
                         :: "v"(ldsOff), "v"(gsrc) : "memory");
        }

        for (int k = 0; k < nv; ++k) {
            // Dense, streaming read of this wave's x row (512B coalesced).
            float4 xv = x4[(size_t)(s0 + k) * F4 + lane];

            if (k + 1 < nv) {
                // Issue next gather before consuming current buffer,
                // then wait until only that new load is outstanding.
                const int lbl = labels[s0 + k + 1];
                const float* gsrc = centers + (size_t)lbl * FEAT + lane * 4;
                unsigned ldsOff =
                    (unsigned)(((wib * 2 + ((k + 1) & 1)) * 32 + lane) * 16);
                asm volatile("global_load_async_to_lds_b128 %0, %1, off"
                             :: "v"(ldsOff), "v"(gsrc) : "memory");
                asm volatile("s_wait_asynccnt 0x1" ::: "memory");
            } else {
                asm volatile("s_wait_asynccnt 0x0" ::: "memory");
            }

            float4 cv = cbuf[(wib * 2 + (k & 1)) * 32 + lane];
            float d0 = xv.x - cv.x;
            float d1 = xv.y - cv.y;
            float d2 = xv.z - cv.z;
            float d3 = xv.w - cv.w;
            acc += d0 * d0 + d1 * d1 + d2 * d2 + d3 * d3;
        }
    }

    // wave32 butterfly reduction (warpSize == 32 on gfx1250).
    #pragma unroll
    for (int m = 16; m > 0; m >>= 1)
        acc += __shfl_xor(acc, m, 32);

    // Cross-wave reduce: reuse cbuf as scratch after all waves are done.
    __syncthreads();                       // everyone finished with cbuf
    float* wsum = (float*)cbuf;
    if (lane == 0) wsum[wib] = acc;
    __syncthreads();

    if (wib == 0) {
        float v = (lane < WAVES_PER_BLOCK) ? wsum[lane] : 0.0f;
        #pragma unroll
        for (int m = WAVES_PER_BLOCK / 2; m > 0; m >>= 1)
            v += __shfl_xor(v, m, WAVES_PER_BLOCK);
        if (lane == 0) partials[blockIdx.x] = v;
    }
}

__global__ __launch_bounds__(32)
void center_finalize(const float* __restrict__ partials, int n,
                     float inv_bc, float* __restrict__ out)
{
    float v = 0.0f;
    for (int i = threadIdx.x; i < n; i += 32)
        v += partials[i];
    #pragma unroll
    for (int m = 16; m > 0; m >>= 1)
        v += __shfl_xor(v, m, 32);
    if (threadIdx.x == 0)
        out[0] = v * inv_bc;
}

extern "C" void kernel_launch(void* const* d_in, const int* in_sizes, int n_in,
                              void* d_out, int out_size, void* d_ws, size_t ws_size,
                              hipStream_t stream)
{
    const float* x       = (const float*)d_in[0];   // (B, 128) fp32
    const int*   labels  = (const int*)d_in[1];     // (B,) int
    const float* centers = (const float*)d_in[2];   // (C, 128) fp32

    const int B = in_sizes[1];
    const int D = in_sizes[0] / B;                  // 128
    const long long C = (long long)(in_sizes[2] / D);

    float* partials = (float*)d_ws;                 // BLOCKS floats of scratch

    const int totalWaves = BLOCKS * WAVES_PER_BLOCK;
    const int spw = (B + totalWaves - 1) / totalWaves;   // samples per wave

    center_gather_partial<<<BLOCKS, THREADS, 0, stream>>>(
        x, labels, centers, partials, B, spw);

    const float inv_bc = (float)(1.0 / ((double)B * (double)C));
    center_finalize<<<1, 32, 0, stream>>>(partials, BLOCKS, inv_bc,
                                          (float*)d_out);
}